// GAT_61409442398711
// MI455X (gfx1250) — compile-verified
//
#include <hip/hip_runtime.h>
#include <hip/hip_bf16.h>
#include <math.h>

typedef __attribute__((ext_vector_type(2))) float v2f;
typedef __attribute__((ext_vector_type(8))) float v8f;

#define NN 50000
#define NE 800000
#define ET (NE + NN)        // edges + self loops = 850000
#define NF 128
#define HID 32
#define HD 8
#define F1 (HD * HID)       // 256
#define NC 16
#define NG 64
#define ENC_NEG_INF ((int)0x807FFFFF)   // enc(-inf)

// ---------- helpers ----------
__device__ __forceinline__ int enc_f(float f) {
    int i = __float_as_int(f);
    return i >= 0 ? i : (i ^ 0x7fffffff);
}
__device__ __forceinline__ float dec_f(int k) {
    int i = k >= 0 ? k : (k ^ 0x7fffffff);
    return __int_as_float(i);
}

__global__ void fill_f32(float* p, float v, long long n) {
    long long t = (long long)blockIdx.x * blockDim.x + threadIdx.x;
    if (t < n) p[t] = v;
}
__global__ void fill_i32(int* p, int v, long long n) {
    long long t = (long long)blockIdx.x * blockDim.x + threadIdx.x;
    if (t < n) p[t] = v;
}

// ---------- dense projection via V_WMMA_F32_16X16X4_F32 ----------
// One wave computes one 16x16 tile of C = A(MxK) * B(KxN). K % 4 == 0, M % 16 == 0.
__global__ void wmma_gemm_f32(const float* __restrict__ A, const float* __restrict__ B,
                              float* __restrict__ C, int Ntiles, int K, int totalTiles) {
    int wave = blockIdx.x * (blockDim.x >> 5) + (threadIdx.x >> 5);
    if (wave >= totalTiles) return;           // wave-uniform: EXEC stays all-1s for WMMA
    int lane = threadIdx.x & 31;
    int tileM = wave / Ntiles;
    int tileN = wave - tileM * Ntiles;
    int N = Ntiles * 16;
    int rsel = lane & 15;
    int kgrp = lane >> 4;                     // 0 or 1

    const float* Arow = A + (size_t)(tileM * 16 + rsel) * K;  // A: lane holds row rsel
    const float* Bcol = B + (size_t)(tileN * 16 + rsel);      // B: lane holds col rsel

    v8f acc = {};
    for (int k = 0; k < K; k += 4) {
        // A 16x4 f32 layout: v0 = K=k+2*kgrp, v1 = K=k+2*kgrp+1
        v2f a = *(const v2f*)(Arow + k + kgrp * 2);
        // B 4x16 f32 layout mirrors A: lanes0-15 rows k..k+1, lanes16-31 rows k+2..k+3
        const float* Bp = Bcol + (size_t)(k + kgrp * 2) * N;
        v2f b;
        b.x = Bp[0];
        b.y = Bp[N];
        acc = __builtin_amdgcn_wmma_f32_16x16x4_f32(
            false, a, false, b, (short)0, acc, false, false);
    }
    // C/D 16x16 f32: lane -> N = rsel ; VGPR r -> M = r + 8*kgrp
    float* Cp = C + (size_t)(tileM * 16 + kgrp * 8) * N + tileN * 16 + rsel;
#pragma unroll
    for (int r = 0; r < 8; ++r) Cp[(size_t)r * N] = acc[r];
}

// ---------- per-node attention coefficients: as[n,h] = <H[n,h,:], a_src[h,:]> ----------
__global__ void alpha_kernel(const float* __restrict__ H, const float* __restrict__ asrc,
                             const float* __restrict__ adst, float* __restrict__ AS,
                             float* __restrict__ AD, int n_nodes, int Hh, int C) {
    long long t = (long long)blockIdx.x * blockDim.x + threadIdx.x;
    if (t >= (long long)n_nodes * Hh) return;
    int n = (int)(t / Hh);
    int h = (int)(t - (long long)n * Hh);
    const float* hp = H + (size_t)n * Hh * C + (size_t)h * C;
    const float* sp = asrc + (size_t)h * C;
    const float* dp = adst + (size_t)h * C;
    float s = 0.f, d = 0.f;
    for (int c = 0; c < C; ++c) {
        float v = hp[c];
        s = fmaf(v, sp[c], s);
        d = fmaf(v, dp[c], d);
    }
    AS[t] = s;
    AD[t] = d;
}

// ---------- edge logits + segment max ----------
__global__ void edge_logits(const float* __restrict__ AS, const float* __restrict__ AD,
                            const int* __restrict__ srcA, const int* __restrict__ dstA,
                            float* __restrict__ E, int* __restrict__ M,
                            int Ereal, long long Etot, int Hh) {
    long long t = (long long)blockIdx.x * blockDim.x + threadIdx.x;
    if (t >= Etot * Hh) return;
    int eid = (int)(t / Hh);
    int h = (int)(t - (long long)eid * Hh);
    int s, d;
    if (eid < Ereal) { s = srcA[eid]; d = dstA[eid]; }
    else             { s = d = eid - Ereal; }
    float e = AS[(size_t)s * Hh + h] + AD[(size_t)d * Hh + h];
    e = e > 0.f ? e : 0.2f * e;               // leaky_relu, slope 0.2
    E[t] = e;
    atomicMax(&M[(size_t)d * Hh + h], enc_f(e));
}

// ---------- softmax numerator + segment sum (E -> P in place) ----------
__global__ void edge_softmax_num(float* __restrict__ E, const int* __restrict__ dstA,
                                 const int* __restrict__ M, float* __restrict__ S,
                                 int Ereal, long long Etot, int Hh) {
    long long t = (long long)blockIdx.x * blockDim.x + threadIdx.x;
    if (t >= Etot * Hh) return;
    int eid = (int)(t / Hh);
    int h = (int)(t - (long long)eid * Hh);
    int d = (eid < Ereal) ? dstA[eid] : (eid - Ereal);
    float m = dec_f(M[(size_t)d * Hh + h]);
    if (!isfinite(m)) m = 0.f;
    float p = expf(E[t] - m);
    E[t] = p;
    atomicAdd(&S[(size_t)d * Hh + h], p);
}

// ---------- weighted aggregation, float4 per thread ----------
// O[dst, f..f+3] += alpha(eid, head(f)) * Hin[src, f..f+3]; 4 | C so a float4
// never straddles a head boundary; 16B alignment holds (F*4 and f*4 are 16B multiples).
__global__ void edge_aggregate_v4(const float* __restrict__ P, const float* __restrict__ S,
                                  const float* __restrict__ Hin, float* __restrict__ O,
                                  const int* __restrict__ srcA, const int* __restrict__ dstA,
                                  int Ereal, long long Etot, int Hh, int C) {
    int F = Hh * C;
    int Fv = F >> 2;                           // float4 groups per edge
    long long t = (long long)blockIdx.x * blockDim.x + threadIdx.x;
    if (t >= Etot * Fv) return;
    int eid = (int)(t / Fv);
    int f = ((int)(t - (long long)eid * Fv)) << 2;
    int h = f / C;
    int s, d;
    if (eid < Ereal) { s = srcA[eid]; d = dstA[eid]; }
    else             { s = d = eid - Ereal; }
    float alpha = P[(size_t)eid * Hh + h] / (S[(size_t)d * Hh + h] + 1e-16f);
    const float4 hv = *(const float4*)(Hin + (size_t)s * F + f);   // global_load_b128 gather
    float* op = O + (size_t)d * F + f;
    atomicAdd(op + 0, alpha * hv.x);
    atomicAdd(op + 1, alpha * hv.y);
    atomicAdd(op + 2, alpha * hv.z);
    atomicAdd(op + 3, alpha * hv.w);
}

__global__ void add_bias(float* __restrict__ O, const float* __restrict__ b,
                         int n_nodes, int F) {
    long long t = (long long)blockIdx.x * blockDim.x + threadIdx.x;
    if (t >= (long long)n_nodes * F) return;
    O[t] += b[t % F];
}

// ---------- global mean pool ----------
__global__ void pool_kernel(const float* __restrict__ O2, const int* __restrict__ batch,
                            float* __restrict__ POOL, float* __restrict__ CNT,
                            int n_nodes, int C) {
    long long t = (long long)blockIdx.x * blockDim.x + threadIdx.x;
    if (t >= (long long)n_nodes * C) return;
    int n = (int)(t / C);
    int c = (int)(t - (long long)n * C);
    int g = batch[n];
    atomicAdd(&POOL[(size_t)g * C + c], O2[t]);
    if (c == 0) atomicAdd(&CNT[g], 1.0f);
}

__global__ void logsoftmax_kernel(const float* __restrict__ POOL, const float* __restrict__ CNT,
                                  float* __restrict__ out, int G, int C) {
    int g = blockIdx.x * blockDim.x + threadIdx.x;
    if (g >= G) return;
    float cnt = CNT[g];
    if (cnt < 1.f) cnt = 1.f;
    float row[NC];
    float mx = -INFINITY;
    for (int c = 0; c < C; ++c) {
        row[c] = POOL[(size_t)g * C + c] / cnt;
        mx = fmaxf(mx, row[c]);
    }
    float se = 0.f;
    for (int c = 0; c < C; ++c) se += expf(row[c] - mx);
    float lse = mx + logf(se);
    for (int c = 0; c < C; ++c) out[(size_t)g * C + c] = row[c] - lse;
}

static inline int cdiv(long long a, int b) { return (int)((a + b - 1) / b); }

extern "C" void kernel_launch(void* const* d_in, const int* in_sizes, int n_in,
                              void* d_out, int out_size, void* d_ws, size_t ws_size,
                              hipStream_t stream) {
    const float* x      = (const float*)d_in[0];
    const int*   ei     = (const int*)d_in[1];
    const int*   batch  = (const int*)d_in[2];
    const float* W1     = (const float*)d_in[3];
    const float* a_src1 = (const float*)d_in[4];
    const float* a_dst1 = (const float*)d_in[5];
    const float* b1     = (const float*)d_in[6];
    const float* W2     = (const float*)d_in[7];
    const float* a_src2 = (const float*)d_in[8];
    const float* a_dst2 = (const float*)d_in[9];
    const float* b2     = (const float*)d_in[10];
    const int* src = ei;
    const int* dst = ei + NE;
    float* out = (float*)d_out;

    // ---- workspace carve-up (256B aligned) ----
    char* w = (char*)d_ws;
    size_t off = 0;
    auto take = [&](size_t bytes) -> void* {
        void* p = w + off;
        off += (bytes + 255) & ~(size_t)255;
        return p;
    };
    float* H1  = (float*)take((size_t)NN * F1 * 4);   // x @ W1
    float* AS1 = (float*)take((size_t)NN * HD * 4);
    float* AD1 = (float*)take((size_t)NN * HD * 4);
    int*   M1  = (int*)  take((size_t)NN * HD * 4);
    float* S1  = (float*)take((size_t)NN * HD * 4);
    float* P1  = (float*)take((size_t)ET * HD * 4);   // logits -> softmax numerators
    float* O1  = (float*)take((size_t)NN * F1 * 4);   // layer-1 output
    float* H2  = (float*)take((size_t)NN * NC * 4);   // O1 @ W2
    float* AS2 = (float*)take((size_t)NN * 4);
    float* AD2 = (float*)take((size_t)NN * 4);
    int*   M2  = (int*)  take((size_t)NN * 4);
    float* S2  = (float*)take((size_t)NN * 4);
    float* P2  = (float*)take((size_t)ET * 4);
    float* O2  = (float*)take((size_t)NN * NC * 4);   // layer-2 output
    float* POOL = (float*)take((size_t)NG * NC * 4);
    float* CNT  = (float*)take((size_t)NG * 4);
    (void)ws_size; (void)in_sizes; (void)n_in; (void)out_size;

    const int T = 256;
    // ---- init accumulators ----
    fill_i32<<<cdiv((long long)NN * HD, T), T, 0, stream>>>(M1, ENC_NEG_INF, (long long)NN * HD);
    fill_f32<<<cdiv((long long)NN * HD, T), T, 0, stream>>>(S1, 0.f, (long long)NN * HD);
    fill_f32<<<cdiv((long long)NN * F1, T), T, 0, stream>>>(O1, 0.f, (long long)NN * F1);
    fill_i32<<<cdiv(NN, T), T, 0, stream>>>(M2, ENC_NEG_INF, NN);
    fill_f32<<<cdiv(NN, T), T, 0, stream>>>(S2, 0.f, NN);
    fill_f32<<<cdiv((long long)NN * NC, T), T, 0, stream>>>(O2, 0.f, (long long)NN * NC);
    fill_f32<<<cdiv(NG * NC, T), T, 0, stream>>>(POOL, 0.f, NG * NC);
    fill_f32<<<cdiv(NG, T), T, 0, stream>>>(CNT, 0.f, NG);

    // ---- layer 1 ----
    {   // H1 = x @ W1 : M=50000 (3125 tiles), N=256 (16 tiles), K=128
        int tiles = 3125 * 16;
        wmma_gemm_f32<<<cdiv(tiles, 4), 128, 0, stream>>>(x, W1, H1, 16, NF, tiles);
    }
    alpha_kernel<<<cdiv((long long)NN * HD, T), T, 0, stream>>>(H1, a_src1, a_dst1, AS1, AD1, NN, HD, HID);
    edge_logits<<<cdiv((long long)ET * HD, T), T, 0, stream>>>(AS1, AD1, src, dst, P1, M1, NE, ET, HD);
    edge_softmax_num<<<cdiv((long long)ET * HD, T), T, 0, stream>>>(P1, dst, M1, S1, NE, ET, HD);
    edge_aggregate_v4<<<cdiv((long long)ET * (F1 / 4), T), T, 0, stream>>>(P1, S1, H1, O1, src, dst, NE, ET, HD, HID);
    add_bias<<<cdiv((long long)NN * F1, T), T, 0, stream>>>(O1, b1, NN, F1);

    // ---- layer 2 ----
    {   // H2 = O1 @ W2 : M=50000 (3125 tiles), N=16 (1 tile), K=256
        int tiles = 3125;
        wmma_gemm_f32<<<cdiv(tiles, 4), 128, 0, stream>>>(O1, W2, H2, 1, F1, tiles);
    }
    alpha_kernel<<<cdiv(NN, T), T, 0, stream>>>(H2, a_src2, a_dst2, AS2, AD2, NN, 1, NC);
    edge_logits<<<cdiv((long long)ET, T), T, 0, stream>>>(AS2, AD2, src, dst, P2, M2, NE, ET, 1);
    edge_softmax_num<<<cdiv((long long)ET, T), T, 0, stream>>>(P2, dst, M2, S2, NE, ET, 1);
    edge_aggregate_v4<<<cdiv((long long)ET * (NC / 4), T), T, 0, stream>>>(P2, S2, H2, O2, src, dst, NE, ET, 1, NC);
    add_bias<<<cdiv((long long)NN * NC, T), T, 0, stream>>>(O2, b2, NN, NC);

    // ---- global mean pool + log_softmax ----
    pool_kernel<<<cdiv((long long)NN * NC, T), T, 0, stream>>>(O2, batch, POOL, CNT, NN, NC);
    logsoftmax_kernel<<<1, 64, 0, stream>>>(POOL, CNT, out, NG, NC);
}